// GINEResidualModel_73632919323006
// MI455X (gfx1250) — compile-verified
//
#include <hip/hip_runtime.h>
#include <hip/hip_bf16.h>

#define NNODES 50000
#define NEDGES 500000
#define DIM    128
#define EDIM   16
#define PITCH  132   // 132 % 64 = 4 -> 4-bank stride, conflict-free strided row reads
#define APITCH 20    // edge_attr tile pitch

typedef float v2f __attribute__((ext_vector_type(2)));
typedef float v8f __attribute__((ext_vector_type(8)));

// ---------------- WMMA helpers (V_WMMA_F32_16X16X4_F32) ----------------
// A 16x4 layout: lanes 0-15 -> M=lane, {K, K+1}; lanes 16-31 -> M=lane-16, {K+2, K+3}
// B 4x16 layout: lanes 0-15 -> N=lane, rows {K, K+1}; lanes 16-31 -> N=lane-16, rows {K+2, K+3}
// C/D 16x16:    VGPR r, lanes 0-15 -> M=r, N=lane; lanes 16-31 -> M=r+8, N=lane-16

__device__ __forceinline__ v8f wmma4(v2f a, v2f b, v8f c) {
  return __builtin_amdgcn_wmma_f32_16x16x4_f32(false, a, false, b, (short)0, c, false, false);
}

// Adjacent-pair A load: single aligned ds_load_b64 (pitch and k are even).
__device__ __forceinline__ v2f ldA(const float* s, int pitch, int mBase, int k, int lane) {
  int m  = mBase + (lane & 15);
  int kk = k + ((lane >> 4) << 1);
  return *(const v2f*)(s + m * pitch + kk);
}

// B staged K-pair interleaved: sB[(p*nCols + n)*2 + {0,1}] = B[2p][n], B[2p+1][n].
// Lanes 0-15 take pair p = k/2; lanes 16-31 take p = k/2 + 1 -> one ds_load_b64.
__device__ __forceinline__ v2f ldB(const float* s, int nCols, int k, int nBase, int lane) {
  int n = nBase + (lane & 15);
  int p = (k >> 1) + (lane >> 4);
  return *(const v2f*)(s + (p * nCols + n) * 2);
}

#define V8ZERO {0.f,0.f,0.f,0.f,0.f,0.f,0.f,0.f}

// ---------------- Kernel 1: edge message + scatter-add ----------------
// proj = edge_attr @ Wᵀ + b (WMMA, K=16); msg = relu(x[src] + proj); agg[dst] += msg
__global__ __launch_bounds__(256) void edge_scatter_kernel(
    const float* __restrict__ x, const int* __restrict__ eidx,
    const float* __restrict__ attr, const float* __restrict__ w,
    const float* __restrict__ bias, float* __restrict__ agg, int nE) {
  __shared__ float sAttr[128 * APITCH];
  __shared__ float sB[EDIM * DIM];       // pair-interleaved: (EDIM/2) x DIM x float2
  __shared__ float sBias[DIM];
  __shared__ float sProj[128 * PITCH];

  const int tid  = threadIdx.x;
  const int lane = tid & 31;
  const int wave = tid >> 5;
  const int tile0 = blockIdx.x * 128;

  // stage edge_attr tile [128 x 16] (zero-pad past nE)
  for (int i = tid; i < 128 * 4; i += 256) {
    int e = i >> 2, q = (i & 3) * 4;
    float4 v = make_float4(0.f, 0.f, 0.f, 0.f);
    int ge = tile0 + e;
    if (ge < nE) v = *(const float4*)(attr + ge * EDIM + q);
    *(float4*)(sAttr + e * APITCH + q) = v;
  }
  // stage B = Wᵀ (pair-interleaved): w is [128 x 16]
  for (int i = tid; i < DIM * EDIM; i += 256) {
    int n = i >> 4, k = i & 15;
    sB[((k >> 1) * DIM + n) * 2 + (k & 1)] = w[n * EDIM + k];
  }
  if (tid < DIM) sBias[tid] = bias[tid];
  __syncthreads();

  const int mBase = wave * 16;
  const int half  = lane >> 4;
  const int nlo   = lane & 15;
  for (int nt = 0; nt < 8; ++nt) {
    v8f acc = V8ZERO;
    #pragma unroll
    for (int k = 0; k < EDIM; k += 4) {
      v2f a = ldA(sAttr, APITCH, mBase, k, lane);
      v2f b = ldB(sB, DIM, k, nt * 16, lane);
      acc = wmma4(a, b, acc);
    }
    int col = nt * 16 + nlo;
    float bv = sBias[col];
    #pragma unroll
    for (int r = 0; r < 8; ++r)
      sProj[(mBase + r + half * 8) * PITCH + col] = acc[r] + bv;
  }
  __syncthreads();

  // scatter: this wave owns edges [mBase, mBase+16)
  for (int i = 0; i < 16; ++i) {
    int ge = tile0 + mBase + i;
    if (ge >= nE) break;                 // uniform across the wave
    int s = eidx[ge];
    int d = eidx[nE + ge];
    int c0 = lane * 4;
    float4 xs = *(const float4*)(x + s * DIM + c0);
    float4 pv = *(const float4*)(&sProj[(mBase + i) * PITCH + c0]);  // 16B-aligned
    float* ar = agg + d * DIM + c0;
    float v0 = fmaxf(xs.x + pv.x, 0.f);
    float v1 = fmaxf(xs.y + pv.y, 0.f);
    float v2 = fmaxf(xs.z + pv.z, 0.f);
    float v3 = fmaxf(xs.w + pv.w, 0.f);
    atomicAdd(ar + 0, v0); atomicAdd(ar + 1, v1);
    atomicAdd(ar + 2, v2); atomicAdd(ar + 3, v3);
  }
}

// ---------------- Kernel 2: node MLP  out = relu((x+agg)W1ᵀ+b1)W2ᵀ+b2 ----------------
__global__ __launch_bounds__(256) void node_mlp_kernel(
    const float* __restrict__ x, const float* __restrict__ agg,
    const float* __restrict__ w1, const float* __restrict__ b1,
    const float* __restrict__ w2, const float* __restrict__ b2,
    float* __restrict__ out, int nN) {
  __shared__ float sA[128 * PITCH];
  __shared__ float sB[DIM * DIM];        // pair-interleaved
  __shared__ float sBias[DIM];

  const int tid  = threadIdx.x;
  const int lane = tid & 31;
  const int wave = tid >> 5;
  const int tile0 = blockIdx.x * 128;
  const int valid = min(128, nN - tile0);

  for (int i = tid; i < 128 * 32; i += 256) {
    int row = i >> 5, c4 = (i & 31) * 4;
    float4 v = make_float4(0.f, 0.f, 0.f, 0.f);
    if (row < valid) {
      int g = (tile0 + row) * DIM + c4;
      float4 xa = *(const float4*)(x + g);
      float4 ag = *(const float4*)(agg + g);
      v = make_float4(xa.x + ag.x, xa.y + ag.y, xa.z + ag.z, xa.w + ag.w);
    }
    *(float4*)(sA + row * PITCH + c4) = v;
  }
  for (int i = tid; i < DIM * DIM; i += 256) {
    int n = i >> 7, k = i & 127;
    sB[((k >> 1) * DIM + n) * 2 + (k & 1)] = w1[n * DIM + k];
  }
  if (tid < DIM) sBias[tid] = b1[tid];
  __syncthreads();

  const int mBase = wave * 16;
  const int half  = lane >> 4;
  const int nlo   = lane & 15;

  v8f acc[8];
  #pragma unroll
  for (int nt = 0; nt < 8; ++nt) {
    v8f c = V8ZERO;
    for (int k = 0; k < DIM; k += 4) {
      v2f a = ldA(sA, PITCH, mBase, k, lane);
      v2f b = ldB(sB, DIM, k, nt * 16, lane);
      c = wmma4(a, b, c);
    }
    acc[nt] = c;
  }
  __syncthreads();  // all reads of sA/sB done

  // relu(acc + b1) back into sA; reload sB with W2ᵀ
  #pragma unroll
  for (int nt = 0; nt < 8; ++nt) {
    int col = nt * 16 + nlo;
    float bv = sBias[col];
    #pragma unroll
    for (int r = 0; r < 8; ++r)
      sA[(mBase + r + half * 8) * PITCH + col] = fmaxf(acc[nt][r] + bv, 0.f);
  }
  for (int i = tid; i < DIM * DIM; i += 256) {
    int n = i >> 7, k = i & 127;
    sB[((k >> 1) * DIM + n) * 2 + (k & 1)] = w2[n * DIM + k];
  }
  __syncthreads();
  if (tid < DIM) sBias[tid] = b2[tid];
  __syncthreads();

  for (int nt = 0; nt < 8; ++nt) {
    v8f c = V8ZERO;
    for (int k = 0; k < DIM; k += 4) {
      v2f a = ldA(sA, PITCH, mBase, k, lane);
      v2f b = ldB(sB, DIM, k, nt * 16, lane);
      c = wmma4(a, b, c);
    }
    int col = nt * 16 + nlo;
    float bv = sBias[col];
    #pragma unroll
    for (int r = 0; r < 8; ++r) {
      int row = mBase + r + half * 8;
      if (row < valid) out[(tile0 + row) * DIM + col] = c[r] + bv;
    }
  }
}

// ---------------- Kernel 3: merge (concat GEMM, K=256 in two halves) + BN stats ----------------
__global__ __launch_bounds__(256) void merge_stats_kernel(
    const float* __restrict__ F, const float* __restrict__ Bm,
    const float* __restrict__ wm, const float* __restrict__ bm,
    float* __restrict__ dest, float* __restrict__ stats, int nN) {
  __shared__ float sA[128 * PITCH];
  __shared__ float sB[DIM * DIM];        // pair-interleaved
  __shared__ float sBias[DIM];

  const int tid  = threadIdx.x;
  const int lane = tid & 31;
  const int wave = tid >> 5;
  const int tile0 = blockIdx.x * 128;
  const int valid = min(128, nN - tile0);
  const int mBase = wave * 16;
  const int half  = lane >> 4;
  const int nlo   = lane & 15;

  v8f acc[8];
  #pragma unroll
  for (int nt = 0; nt < 8; ++nt) acc[nt] = (v8f)V8ZERO;

  for (int hf = 0; hf < 2; ++hf) {
    const float* src = hf ? Bm : F;
    for (int i = tid; i < 128 * 32; i += 256) {
      int row = i >> 5, c4 = (i & 31) * 4;
      float4 v = make_float4(0.f, 0.f, 0.f, 0.f);
      if (row < valid) v = *(const float4*)(src + (tile0 + row) * DIM + c4);
      *(float4*)(sA + row * PITCH + c4) = v;
    }
    for (int i = tid; i < DIM * DIM; i += 256) {
      int n = i >> 7, k = i & 127;
      sB[((k >> 1) * DIM + n) * 2 + (k & 1)] = wm[n * 2 * DIM + hf * DIM + k];
    }
    if (hf == 0 && tid < DIM) sBias[tid] = bm[tid];
    __syncthreads();
    #pragma unroll
    for (int nt = 0; nt < 8; ++nt) {
      v8f c = acc[nt];
      for (int k = 0; k < DIM; k += 4) {
        v2f a = ldA(sA, PITCH, mBase, k, lane);
        v2f b = ldB(sB, DIM, k, nt * 16, lane);
        c = wmma4(a, b, c);
      }
      acc[nt] = c;
    }
    __syncthreads();  // reads done before next half overwrites (and before H store)
  }

  // H = acc + bias -> dest and sA (for stats)
  #pragma unroll
  for (int nt = 0; nt < 8; ++nt) {
    int col = nt * 16 + nlo;
    float bv = sBias[col];
    #pragma unroll
    for (int r = 0; r < 8; ++r) {
      int row = mBase + r + half * 8;
      float v = acc[nt][r] + bv;
      sA[row * PITCH + col] = v;
      if (row < valid) dest[(tile0 + row) * DIM + col] = v;
    }
  }
  __syncthreads();

  if (tid < DIM) {
    float s = 0.f, sq = 0.f;
    for (int r = 0; r < valid; ++r) {
      float v = sA[r * PITCH + tid];
      s += v; sq += v * v;
    }
    atomicAdd(&stats[tid], s);
    atomicAdd(&stats[DIM + tid], sq);
  }
}

// ---------------- Kernel 4: BN (training stats) + leaky ReLU, in place ----------------
__global__ __launch_bounds__(256) void bn_lrelu_kernel(
    float* __restrict__ h, const float* __restrict__ stats,
    const float* __restrict__ gamma, const float* __restrict__ beta, int nN) {
  int idx = blockIdx.x * 256 + threadIdx.x;
  int total = nN * DIM;
  if (idx >= total) return;
  int c = idx & (DIM - 1);
  float invN = 1.0f / (float)nN;
  float mu  = stats[c] * invN;
  float var = stats[DIM + c] * invN - mu * mu;
  float g = gamma[c] * rsqrtf(var + 1e-5f);
  float v = (h[idx] - mu) * g + beta[c];
  h[idx] = (v > 0.f) ? v : 0.01f * v;
}

// ---------------- Launch ----------------
extern "C" void kernel_launch(void* const* d_in, const int* in_sizes, int n_in,
                              void* d_out, int out_size, void* d_ws, size_t ws_size,
                              hipStream_t stream) {
  (void)in_sizes; (void)n_in; (void)out_size; (void)ws_size;
  const float* x0   = (const float*)d_in[0];
  const int*   fwdE = (const int*)d_in[1];
  const int*   bwdE = (const int*)d_in[2];
  const float* attr = (const float*)d_in[3];
  auto P = [&](int l, int off) { return (const float*)d_in[4 + l * 16 + off]; };
  // per-layer leaves: 0 fwd.edge_lin.w 1 .b  2 fwd.mlp1.w 3 .b  4 fwd.mlp2.w 5 .b
  //                   6 bwd.edge_lin.w 7 .b  8 bwd.mlp1.w 9 .b 10 bwd.mlp2.w 11 .b
  //                  12 merge.w 13 merge.b  14 bn.gamma 15 bn.beta

  const size_t mat = (size_t)NNODES * DIM * sizeof(float);
  char* ws = (char*)d_ws;
  float* X1    = (float*)(ws);
  float* AGG   = (float*)(ws + mat);
  float* Fb    = (float*)(ws + 2 * mat);
  float* Bb    = (float*)(ws + 3 * mat);
  float* stats = (float*)(ws + 4 * mat);

  const int EB = (NEDGES + 127) / 128;       // 3907
  const int NB = (NNODES + 127) / 128;       // 391
  const int PB = (NNODES * DIM + 255) / 256; // 25000

  for (int l = 0; l < 2; ++l) {
    const float* Xin = l ? (const float*)X1 : x0;
    float* Dest = l ? (float*)d_out : X1;

    hipMemsetAsync(AGG, 0, mat, stream);
    edge_scatter_kernel<<<EB, 256, 0, stream>>>(Xin, fwdE, attr, P(l, 0), P(l, 1), AGG, NEDGES);
    node_mlp_kernel<<<NB, 256, 0, stream>>>(Xin, AGG, P(l, 2), P(l, 3), P(l, 4), P(l, 5), Fb, NNODES);

    hipMemsetAsync(AGG, 0, mat, stream);
    edge_scatter_kernel<<<EB, 256, 0, stream>>>(Xin, bwdE, attr, P(l, 6), P(l, 7), AGG, NEDGES);
    node_mlp_kernel<<<NB, 256, 0, stream>>>(Xin, AGG, P(l, 8), P(l, 9), P(l, 10), P(l, 11), Bb, NNODES);

    hipMemsetAsync(stats, 0, 2 * DIM * sizeof(float), stream);
    merge_stats_kernel<<<NB, 256, 0, stream>>>(Fb, Bb, P(l, 12), P(l, 13), Dest, stats, NNODES);
    bn_lrelu_kernel<<<PB, 256, 0, stream>>>(Dest, stats, P(l, 14), P(l, 15), NNODES);
  }
}